// GAT_13589276524661
// MI455X (gfx1250) — compile-verified
//
#include <hip/hip_runtime.h>

#define N_NODES 50000
#define N_EDGES 800000
#define E_TOT   (N_EDGES + N_NODES)   // 850000, self-loops appended
#define IN_F    128
#define HID     64
#define HEADS   4
#define NH      (HEADS * HID)         // 256
#define SLOPE   0.2f

typedef float v2f __attribute__((ext_vector_type(2)));
typedef float v8f __attribute__((ext_vector_type(8)));

// ---- monotone float<->int encoding for integer atomicMax on floats ----
__device__ __forceinline__ int enc_f(float f) {
  int i = __float_as_int(f);
  return i >= 0 ? i : (i ^ 0x7fffffff);
}
__device__ __forceinline__ float dec_f(int i) {
  return __int_as_float(i >= 0 ? i : (i ^ 0x7fffffff));
}
#define ENC_NEG_INF ((int)0x807fffff)  // enc(-inf)

__device__ __forceinline__ void edge_sd(const int* __restrict__ ei, int e, int& s, int& d) {
  if (e < N_EDGES) { s = ei[e]; d = ei[N_EDGES + e]; }
  else             { s = e - N_EDGES; d = e - N_EDGES; }
}

__device__ __forceinline__ float lrelu(float x) { return x > 0.f ? x : x * SLOPE; }

// ---------------------------------------------------------------------------
// Workspace init: zero accumulators, set segment-max slots to enc(-inf)
// ---------------------------------------------------------------------------
__global__ __launch_bounds__(256) void init_ws_kernel(
    float* __restrict__ accum1, int* __restrict__ emax1, float* __restrict__ denom1,
    float* __restrict__ accum2, int* __restrict__ emax2, float* __restrict__ denom2) {
  int i = blockIdx.x * 256 + threadIdx.x;
  if (i < N_NODES * NH) accum1[i] = 0.f;
  if (i < N_NODES * HEADS) { emax1[i] = ENC_NEG_INF; denom1[i] = 0.f; }
  if (i < N_NODES) { accum2[i] = 0.f; emax2[i] = ENC_NEG_INF; denom2[i] = 0.f; }
}

// ---------------------------------------------------------------------------
// Layer-1 GEMM: h[50000,256] = x[50000,128] @ W1[128,256], fp32 WMMA.
// grid = (3125, 4), block = 128 (4 waves). Wave w -> C tile (blockIdx.x, 4*y+w).
// ---------------------------------------------------------------------------
__global__ __launch_bounds__(128) void gemm1_wmma(
    const float* __restrict__ x, const float* __restrict__ W, float* __restrict__ h) {
  const int wave = threadIdx.x >> 5;
  const int lane = threadIdx.x & 31;
  const int half = lane >> 4;         // 0: K lo pair, 1: K hi pair
  const int r    = lane & 15;
  const int mtile = blockIdx.x;
  const int ntile = blockIdx.y * 4 + wave;

  const float* __restrict__ arow = x + (size_t)(mtile * 16 + r) * IN_F + 2 * half;
  const float* __restrict__ bcol = W + (size_t)(2 * half) * NH + (ntile * 16 + r);

  v8f acc = {};
  #pragma unroll 4
  for (int k = 0; k < IN_F; k += 4) {
    v2f a = *(const v2f*)arow;        // A[m][k+2h], A[m][k+2h+1]  (global_load_b64)
    v2f b;
    b.x = bcol[0];                    // B[k+2h][n]
    b.y = bcol[NH];                   // B[k+2h+1][n]
    acc = __builtin_amdgcn_wmma_f32_16x16x4_f32(
        /*neg_a=*/false, a, /*neg_b=*/false, b,
        /*c_mod=*/(short)0, acc, /*reuse_a=*/false, /*reuse_b=*/false);
    arow += 4;
    bcol += 4 * NH;
  }

  // C layout: VGPR v -> (lanes 0-15: M=v, N=lane) (lanes 16-31: M=8+v, N=lane-16)
  const int row0 = mtile * 16 + half * 8;
  const int col  = ntile * 16 + r;
  float* __restrict__ out = h + (size_t)row0 * NH + col;
  #pragma unroll
  for (int v = 0; v < 8; ++v) out[(size_t)v * NH] = acc[v];
}

// ---------------------------------------------------------------------------
// a_src[n][h] = <h_row[h], att_src1[h]>, same for a_dst. One wave per node.
// ---------------------------------------------------------------------------
__global__ __launch_bounds__(256) void att_reduce1(
    const float* __restrict__ h, const float* __restrict__ att_s,
    const float* __restrict__ att_d, float* __restrict__ as1, float* __restrict__ ad1) {
  const int node = blockIdx.x * 8 + (threadIdx.x >> 5);
  if (node >= N_NODES) return;
  const int lane = threadIdx.x & 31;
  const int head = lane >> 3;         // 4 heads x 8 lanes
  const int sub  = lane & 7;          // each lane covers 8 channels
  const float* __restrict__ hp  = h     + (size_t)node * NH + head * HID + sub * 8;
  const float* __restrict__ asp = att_s + head * HID + sub * 8;
  const float* __restrict__ adp = att_d + head * HID + sub * 8;
  float s = 0.f, d = 0.f;
  #pragma unroll
  for (int j = 0; j < 8; ++j) { float v = hp[j]; s += v * asp[j]; d += v * adp[j]; }
  #pragma unroll
  for (int off = 1; off < 8; off <<= 1) {
    s += __shfl_xor(s, off, 32);
    d += __shfl_xor(d, off, 32);
  }
  if (sub == 0) { as1[node * HEADS + head] = s; ad1[node * HEADS + head] = d; }
}

// ---------------------------------------------------------------------------
// Layer-1 edge softmax pass 1: segment max over dst (int-encoded atomicMax)
// ---------------------------------------------------------------------------
__global__ __launch_bounds__(256) void edge_max1(
    const int* __restrict__ ei, const float* __restrict__ as1,
    const float* __restrict__ ad1, int* __restrict__ emax1) {
  int idx = blockIdx.x * 256 + threadIdx.x;
  if (idx >= E_TOT * HEADS) return;
  int e = idx >> 2, hd = idx & 3;
  int s, d; edge_sd(ei, e, s, d);
  float le = lrelu(as1[s * HEADS + hd] + ad1[d * HEADS + hd]);
  atomicMax(&emax1[d * HEADS + hd], enc_f(le));
}

// Pass 2: denom[dst][h] += exp(le - max)
__global__ __launch_bounds__(256) void edge_denom1(
    const int* __restrict__ ei, const float* __restrict__ as1,
    const float* __restrict__ ad1, const int* __restrict__ emax1,
    float* __restrict__ denom1) {
  int idx = blockIdx.x * 256 + threadIdx.x;
  if (idx >= E_TOT * HEADS) return;
  int e = idx >> 2, hd = idx & 3;
  int s, d; edge_sd(ei, e, s, d);
  float le = lrelu(as1[s * HEADS + hd] + ad1[d * HEADS + hd]);
  float m = dec_f(emax1[d * HEADS + hd]);
  atomicAdd(&denom1[d * HEADS + hd], expf(le - m));
}

// Pass 3: accum[dst][h*64+c] += h[src][h*64+c] * alpha.  One wave per edge.
__global__ __launch_bounds__(256) void edge_scatter1(
    const int* __restrict__ ei, const float* __restrict__ h,
    const float* __restrict__ as1, const float* __restrict__ ad1,
    const int* __restrict__ emax1, const float* __restrict__ denom1,
    float* __restrict__ accum1) {
  const int e = blockIdx.x * 8 + (threadIdx.x >> 5);
  if (e >= E_TOT) return;
  const int lane = threadIdx.x & 31;
  int s, d; edge_sd(ei, e, s, d);
  float alpha[HEADS];
  #pragma unroll
  for (int hd = 0; hd < HEADS; ++hd) {
    float le = lrelu(as1[s * HEADS + hd] + ad1[d * HEADS + hd]);
    float m  = dec_f(emax1[d * HEADS + hd]);
    alpha[hd] = expf(le - m) / denom1[d * HEADS + hd];
  }
  const float* __restrict__ hs = h + (size_t)s * NH;
  float* __restrict__ acc = accum1 + (size_t)d * NH;
  #pragma unroll
  for (int j = 0; j < 8; ++j) {
    int c = lane + 32 * j;
    atomicAdd(&acc[c], hs[c] * alpha[c >> 6]);
  }
}

// ---------------------------------------------------------------------------
// Finalize layer 1 (head mean + bias + ELU) and fold in layer-2 projection:
// h2[n] = elu_row @ W2; a2s = h2*att_src2; a2d = h2*att_dst2. One wave/node.
// ---------------------------------------------------------------------------
__global__ __launch_bounds__(256) void finalize1(
    const float* __restrict__ accum1, const float* __restrict__ b1,
    const float* __restrict__ W2, const float* __restrict__ att_s2,
    const float* __restrict__ att_d2, float* __restrict__ h2,
    float* __restrict__ a2s, float* __restrict__ a2d) {
  const int node = blockIdx.x * 8 + (threadIdx.x >> 5);
  if (node >= N_NODES) return;
  const int lane = threadIdx.x & 31;
  const float* __restrict__ a = accum1 + (size_t)node * NH;
  float part = 0.f;
  #pragma unroll
  for (int j = 0; j < 2; ++j) {
    int c = lane + 32 * j;
    float s = 0.25f * (a[c] + a[HID + c] + a[2 * HID + c] + a[3 * HID + c]) + b1[c];
    float x1 = s > 0.f ? s : (expf(s) - 1.f);   // ELU
    part += x1 * W2[c];
  }
  #pragma unroll
  for (int off = 16; off; off >>= 1) part += __shfl_xor(part, off, 32);
  if (lane == 0) {
    h2[node]  = part;
    a2s[node] = part * att_s2[0];
    a2d[node] = part * att_d2[0];
  }
}

// ---------------------------------------------------------------------------
// Layer-2 edge softmax (1 head, 1 channel)
// ---------------------------------------------------------------------------
__global__ __launch_bounds__(256) void edge_max2(
    const int* __restrict__ ei, const float* __restrict__ a2s,
    const float* __restrict__ a2d, int* __restrict__ emax2) {
  int e = blockIdx.x * 256 + threadIdx.x;
  if (e >= E_TOT) return;
  int s, d; edge_sd(ei, e, s, d);
  atomicMax(&emax2[d], enc_f(lrelu(a2s[s] + a2d[d])));
}

__global__ __launch_bounds__(256) void edge_denom2(
    const int* __restrict__ ei, const float* __restrict__ a2s,
    const float* __restrict__ a2d, const int* __restrict__ emax2,
    float* __restrict__ denom2) {
  int e = blockIdx.x * 256 + threadIdx.x;
  if (e >= E_TOT) return;
  int s, d; edge_sd(ei, e, s, d);
  float le = lrelu(a2s[s] + a2d[d]);
  atomicAdd(&denom2[d], expf(le - dec_f(emax2[d])));
}

__global__ __launch_bounds__(256) void edge_scatter2(
    const int* __restrict__ ei, const float* __restrict__ a2s,
    const float* __restrict__ a2d, const int* __restrict__ emax2,
    const float* __restrict__ denom2, const float* __restrict__ h2,
    float* __restrict__ accum2) {
  int e = blockIdx.x * 256 + threadIdx.x;
  if (e >= E_TOT) return;
  int s, d; edge_sd(ei, e, s, d);
  float le = lrelu(a2s[s] + a2d[d]);
  float alpha = expf(le - dec_f(emax2[d])) / denom2[d];
  atomicAdd(&accum2[d], h2[s] * alpha);
}

__global__ __launch_bounds__(256) void final_out(
    const float* __restrict__ accum2, const float* __restrict__ b2,
    float* __restrict__ out) {
  int n = blockIdx.x * 256 + threadIdx.x;
  if (n < N_NODES) out[n] = accum2[n] + b2[0];
}

// ---------------------------------------------------------------------------
extern "C" void kernel_launch(void* const* d_in, const int* in_sizes, int n_in,
                              void* d_out, int out_size, void* d_ws, size_t ws_size,
                              hipStream_t stream) {
  const float* x       = (const float*)d_in[0];
  const int*   ei      = (const int*)d_in[1];   // [2, 800000] edge_index
  // d_in[2] = edge_attr (unused by reference)
  const float* W1      = (const float*)d_in[3];
  const float* att_s1  = (const float*)d_in[4];
  const float* att_d1  = (const float*)d_in[5];
  const float* b1      = (const float*)d_in[6];
  const float* W2      = (const float*)d_in[7];
  const float* att_s2  = (const float*)d_in[8];
  const float* att_d2  = (const float*)d_in[9];
  const float* b2      = (const float*)d_in[10];
  float*       out     = (float*)d_out;

  // Workspace layout (all 4B types; base is allocation-aligned, h first for
  // 8B-aligned float2 WMMA-adjacent loads).
  char* p = (char*)d_ws;
  float* h      = (float*)p; p += (size_t)N_NODES * NH * 4;       // 51.2 MB
  float* accum1 = (float*)p; p += (size_t)N_NODES * NH * 4;       // 51.2 MB
  float* as1    = (float*)p; p += (size_t)N_NODES * HEADS * 4;
  float* ad1    = (float*)p; p += (size_t)N_NODES * HEADS * 4;
  int*   emax1  = (int*)p;   p += (size_t)N_NODES * HEADS * 4;
  float* denom1 = (float*)p; p += (size_t)N_NODES * HEADS * 4;
  float* h2     = (float*)p; p += (size_t)N_NODES * 4;
  float* a2s    = (float*)p; p += (size_t)N_NODES * 4;
  float* a2d    = (float*)p; p += (size_t)N_NODES * 4;
  int*   emax2  = (int*)p;   p += (size_t)N_NODES * 4;
  float* denom2 = (float*)p; p += (size_t)N_NODES * 4;
  float* accum2 = (float*)p; p += (size_t)N_NODES * 4;

  // 0) init accumulators / segment-max slots
  init_ws_kernel<<<(N_NODES * NH + 255) / 256, 256, 0, stream>>>(
      accum1, emax1, denom1, accum2, emax2, denom2);

  // 1) h = x @ W1 via fp32 WMMA (3125 M-tiles x 16 N-tiles, 4 waves/block)
  gemm1_wmma<<<dim3(N_NODES / 16, HEADS), 128, 0, stream>>>(x, W1, h);

  // 2) per-node attention logits
  att_reduce1<<<(N_NODES + 7) / 8, 256, 0, stream>>>(h, att_s1, att_d1, as1, ad1);

  // 3-5) layer-1 edge softmax + message scatter
  edge_max1   <<<(E_TOT * HEADS + 255) / 256, 256, 0, stream>>>(ei, as1, ad1, emax1);
  edge_denom1 <<<(E_TOT * HEADS + 255) / 256, 256, 0, stream>>>(ei, as1, ad1, emax1, denom1);
  edge_scatter1<<<(E_TOT + 7) / 8, 256, 0, stream>>>(ei, h, as1, ad1, emax1, denom1, accum1);

  // 6) head mean + bias + ELU, fused with layer-2 projection
  finalize1<<<(N_NODES + 7) / 8, 256, 0, stream>>>(
      accum1, b1, W2, att_s2, att_d2, h2, a2s, a2d);

  // 7-9) layer-2 edge softmax + scalar scatter
  edge_max2   <<<(E_TOT + 255) / 256, 256, 0, stream>>>(ei, a2s, a2d, emax2);
  edge_denom2 <<<(E_TOT + 255) / 256, 256, 0, stream>>>(ei, a2s, a2d, emax2, denom2);
  edge_scatter2<<<(E_TOT + 255) / 256, 256, 0, stream>>>(ei, a2s, a2d, emax2, denom2, h2, accum2);

  // 10) final bias
  final_out<<<(N_NODES + 255) / 256, 256, 0, stream>>>(accum2, b2, out);
}